// KWConv2D_17806934409860
// MI455X (gfx1250) — compile-verified
//
#include <hip/hip_runtime.h>
#include <hip/hip_bf16.h>
#include <stdint.h>

// ---------------------------------------------------------------------------
// Problem constants (from reference)
// ---------------------------------------------------------------------------
#define BATCH   8
#define CI      64
#define HH      256
#define WW      256
#define CO      128
#define NK      8      // N_KERNELS
#define KT      3      // kernel size
#define TAPS    9      // KT*KT
#define KTOT    (TAPS * CI)   // 576 = GEMM K, order k = tap*64 + i
#define NT      64     // pixel tile per block (one image row segment)
#define HW      (HH * WW)

typedef __attribute__((ext_vector_type(2))) float v2f;
typedef __attribute__((ext_vector_type(8))) float v8f;

// ---------------------------------------------------------------------------
// 1) Global average pool: pooled[b][c] = mean over HxW.  One block per (b,c).
// ---------------------------------------------------------------------------
__global__ __launch_bounds__(256) void pool_kernel(const float* __restrict__ x,
                                                   float* __restrict__ pooled) {
    const int bc  = blockIdx.x;                 // 0..511
    const int tid = threadIdx.x;
    const float4* p4 = reinterpret_cast<const float4*>(x + (size_t)bc * HW);
    float s = 0.f;
    for (int e = tid; e < HW / 4; e += 256) {
        float4 v = p4[e];
        s += v.x + v.y + v.z + v.w;
    }
    __shared__ float red[256];
    red[tid] = s;
    __syncthreads();
    for (int off = 128; off > 0; off >>= 1) {
        if (tid < off) red[tid] += red[tid + off];
        __syncthreads();
    }
    if (tid == 0) pooled[bc] = red[0] * (1.0f / (float)HW);
}

// ---------------------------------------------------------------------------
// 2) Attention logits + softmax.  Single block, 64 threads: t = b*8 + n.
// ---------------------------------------------------------------------------
__global__ __launch_bounds__(64) void attn_kernel(const float* __restrict__ pooled,
                                                  const float* __restrict__ attn_w,
                                                  const float* __restrict__ attn_b,
                                                  float* __restrict__ attn) {
    const int t = threadIdx.x;        // 0..63
    const int b = t >> 3;
    const int n = t & 7;
    float lg = attn_b[n];
    for (int c = 0; c < CI; ++c)
        lg += pooled[b * CI + c] * attn_w[n * CI + c];
    __shared__ float L[BATCH * NK];
    L[t] = lg;
    __syncthreads();
    float mx = -3.402823466e+38f;
    for (int j = 0; j < NK; ++j) mx = fmaxf(mx, L[b * NK + j]);
    float den = 0.f;
    for (int j = 0; j < NK; ++j) den += __expf(L[b * NK + j] - mx);
    attn[t] = __expf(lg - mx) / den;
}

// ---------------------------------------------------------------------------
// 3) Kernel mix into GEMM-friendly layout:
//    mixed[b][m][tap][i]  (k = tap*64 + i)   from bank[n][m][i][kh][kw]
// ---------------------------------------------------------------------------
__global__ __launch_bounds__(256) void mix_kernel(const float* __restrict__ bank,
                                                  const float* __restrict__ attn,
                                                  float* __restrict__ mixed) {
    const int idx = blockIdx.x * 256 + threadIdx.x;     // < 8*128*9*64
    const int total = BATCH * CO * TAPS * CI;
    if (idx >= total) return;
    const int i   = idx & 63;
    const int tap = (idx >> 6) % TAPS;
    const int m   = (idx / (TAPS * CI)) % CO;
    const int b   = idx / (TAPS * CI * CO);
    float s = 0.f;
    #pragma unroll
    for (int n = 0; n < NK; ++n)
        s += attn[b * NK + n] * bank[((size_t)(n * CO + m) * CI + i) * TAPS + tap];
    mixed[idx] = s;
}

// ---------------------------------------------------------------------------
// 4) Implicit-GEMM 3x3 conv via V_WMMA_F32_16X16X4_F32 (wave32).
//    Block: 256 thr = 8 waves.  Covers M=128 (all C_OUT) x N=64 pixels of one
//    image row.  Wave w: mgroup = w&3 -> M-tiles {2g, 2g+1} (rows 32g..32g+31),
//    nhalf = w>>2 -> N-tiles {2h, 2h+1} (pixels 32h..32h+31).  2x2 register
//    tile => A and B fragments each reused twice per WMMA quad.
//    K = 576 walked in chunks of 4 (one input-channel quad of one tap).
//    LDS slab: x[b][*][y-1..y+1][xbase-1..xbase+64] as [kh][col][ci], staged
//    with GLOBAL_LOAD_ASYNC_TO_LDS_B32 (ASYNCcnt); pad borders via ds stores.
// ---------------------------------------------------------------------------
#define SLAB_COLS 66
#define SLAB_ELEMS (KT * SLAB_COLS * CI)   // 12672 floats = 50688 B

__global__ __launch_bounds__(256) void conv_kernel(const float* __restrict__ x,
                                                   const float* __restrict__ mixed,
                                                   float* __restrict__ out) {
    __shared__ float slab[SLAB_ELEMS];

    const int bid = blockIdx.x;            // 8 * 256 * 4 = 8192 blocks
    const int b     = bid >> 10;
    const int y     = (bid >> 2) & 255;
    const int xbase = (bid & 3) * NT;
    const int tid   = threadIdx.x;

    // ---- stage x slab: async copy global->LDS for in-range elements, ----
    // ---- plain LDS stores of zero for the padded border (disjoint).   ----
    const float* xb = x + (size_t)b * CI * HW;
    for (int e = tid; e < SLAB_ELEMS; e += 256) {
        const int i   = e / (KT * SLAB_COLS);
        const int rem = e % (KT * SLAB_COLS);
        const int kh  = rem / SLAB_COLS;
        const int col = rem % SLAB_COLS;
        const int yy  = y + kh - 1;
        const int xx  = xbase + col - 1;
        const int d   = (kh * SLAB_COLS + col) * CI + i;
        if (yy >= 0 && yy < HH && xx >= 0 && xx < WW) {
            const unsigned ldsAddr = (unsigned)(uintptr_t)(&slab[d]);
            const unsigned gOff    = (unsigned)(((i * HW) + yy * WW + xx) * 4);
            asm volatile("global_load_async_to_lds_b32 %0, %1, %2"
                         :
                         : "v"(ldsAddr), "v"(gOff), "s"(xb)
                         : "memory");
        } else {
            slab[d] = 0.f;
        }
    }
    asm volatile("s_wait_asynccnt 0x0" ::: "memory");
    __syncthreads();

    const int wave     = tid >> 5;
    const int lane     = tid & 31;
    const int laneHalf = lane >> 4;     // 0: lanes 0-15, 1: lanes 16-31
    const int lane16   = lane & 15;
    const int mg       = wave & 3;      // M-group: rows 32*mg .. 32*mg+31
    const int nh       = wave >> 2;     // N-half:  pixels 32*nh .. 32*nh+31

    // A fragments: A[m][k] row-major; lane loads K pair {k+2h, k+2h+1}
    // (ISA 32-bit A 16x4 layout).  Two M-tiles per wave.
    const float* Arow0 = mixed + ((size_t)(b * CO + mg * 32 + lane16)) * KTOT
                               + 2 * laneHalf;
    const float* Arow1 = Arow0 + (size_t)16 * KTOT;

    v8f acc00 = {}, acc01 = {}, acc10 = {}, acc11 = {};

    for (int k = 0; k < KTOT; k += 4) {
        const int tap = k >> 6;          // uniform
        const int i0  = k & 63;
        const int kh  = tap / 3;
        const int kw  = tap % 3;

        v2f a0 = *(const v2f*)(Arow0 + k);
        v2f a1 = *(const v2f*)(Arow1 + k);

        // B fragment per N-tile: B[k+kr][n] = slab[kh][n+kw][i0+kr];
        // lane holds rows {k+2h, k+2h+1} at column n -> 2 consecutive floats.
        const int colBase = kh * SLAB_COLS + kw + nh * 32 + lane16;
        const float* bbase = &slab[colBase * CI + i0 + 2 * laneHalf];

        v2f b0 = *(const v2f*)(bbase);
        v2f b1 = *(const v2f*)(bbase + 16 * CI);

        acc00 = __builtin_amdgcn_wmma_f32_16x16x4_f32(false, a0, false, b0,
                                                      (short)0, acc00, false, false);
        acc01 = __builtin_amdgcn_wmma_f32_16x16x4_f32(false, a0, false, b1,
                                                      (short)0, acc01, false, false);
        acc10 = __builtin_amdgcn_wmma_f32_16x16x4_f32(false, a1, false, b0,
                                                      (short)0, acc10, false, false);
        acc11 = __builtin_amdgcn_wmma_f32_16x16x4_f32(false, a1, false, b1,
                                                      (short)0, acc11, false, false);
    }

    // ---- store D tiles: lane l, vgpr r holds D[m = mtile*16 + r + 8*half][n] ----
    float* ob = out + (size_t)b * CO * HW + (size_t)y * WW;
    #pragma unroll
    for (int mt = 0; mt < 2; ++mt) {
        const int mBase = mg * 32 + mt * 16 + 8 * laneHalf;
        #pragma unroll
        for (int nt = 0; nt < 2; ++nt) {
            const v8f acc = (mt == 0) ? (nt == 0 ? acc00 : acc01)
                                      : (nt == 0 ? acc10 : acc11);
            const int n = xbase + nh * 32 + nt * 16 + lane16;
            #pragma unroll
            for (int r = 0; r < 8; ++r)
                ob[(size_t)(mBase + r) * HW + n] = acc[r];
        }
    }
}

// ---------------------------------------------------------------------------
// launch
// ---------------------------------------------------------------------------
extern "C" void kernel_launch(void* const* d_in, const int* in_sizes, int n_in,
                              void* d_out, int out_size, void* d_ws, size_t ws_size,
                              hipStream_t stream) {
    const float* x      = (const float*)d_in[0];
    const float* bank   = (const float*)d_in[1];
    const float* attn_w = (const float*)d_in[2];
    const float* attn_b = (const float*)d_in[3];
    float* out = (float*)d_out;

    char* ws = (char*)d_ws;
    float* pooled = (float*)(ws);                 // 512 f32
    float* attn   = (float*)(ws + 2048);          // 64 f32
    float* mixed  = (float*)(ws + 4096);          // 8*128*9*64 f32 = 2.36 MB

    pool_kernel<<<BATCH * CI, 256, 0, stream>>>(x, pooled);
    attn_kernel<<<1, 64, 0, stream>>>(pooled, attn_w, attn_b, attn);

    const int mixTotal = BATCH * CO * TAPS * CI;
    mix_kernel<<<(mixTotal + 255) / 256, 256, 0, stream>>>(bank, attn, mixed);

    conv_kernel<<<BATCH * HH * (WW / NT), 256, 0, stream>>>(x, mixed, out);
}